// HNN_4741643895138
// MI455X (gfx1250) — compile-verified
//
#include <hip/hip_runtime.h>
#include <math.h>

// ---------------------------------------------------------------------------
// HGT forward for MI455X (gfx1250, wave32).
// Dense node GEMMs use V_WMMA_F32_16X16X4_F32 (fp32 in/accum == reference).
// Relation transforms a_rel/m_rel are folded into Wk/Wv (node-side), removing
// all per-edge GEMMs. Edge phase = coalesced gathers + f32 atomics (BW bound).
// GEMM is templated on the output width so all B addressing folds into
// compile-time immediate offsets (no per-iteration 64-bit address VALU).
// ---------------------------------------------------------------------------

#define HID 128
#define NH  8
#define HD  16

typedef __attribute__((ext_vector_type(2))) float v2f;
typedef __attribute__((ext_vector_type(8))) float v8f;

__device__ __forceinline__ float gelu_exact(float x) {
  return 0.5f * x * (1.0f + erff(x * 0.70710678118654752f));
}

// float atomic max via sign-aware integer atomics (dest initialized to -inf)
__device__ __forceinline__ void atomicMaxFloat(float* addr, float val) {
  if (val >= 0.0f)
    atomicMax((int*)addr, __float_as_int(val));
  else
    atomicMin((unsigned int*)addr, __float_as_uint(val));
}

__global__ void fill_kernel(float* __restrict__ p, float v, int n) {
  int i = blockIdx.x * blockDim.x + threadIdx.x;
  if (i < n) p[i] = v;
}

// ---------------------------------------------------------------------------
// C[M,NC] = epilogue( f(A[M,128]) @ B[128,NC] + bias )
//   f = exact GELU if geluIn, epilogue = ReLU if reluOut,
//   gated residual if (resid,skipP): C = sigma(skip)*C + (1-sigma(skip))*resid
// Block: 2*NC threads (one wave per 16-col tile), grid: ceil(M/16).
// NC is a template parameter so B offsets are compile-time immediates.
// ---------------------------------------------------------------------------
template <int NC>
__global__ __launch_bounds__(256) void gemm_wmma_kernel(
    const float* __restrict__ A, const float* __restrict__ B,
    const float* __restrict__ bias, const float* __restrict__ resid,
    const float* __restrict__ skipP, float* __restrict__ C,
    int M, int geluIn, int reluOut)
{
  __shared__ float As[16 * 132];          // 132-stride: conflict-free column reads
  const int tid  = threadIdx.x;
  const int wave = tid >> 5;
  const int lane = tid & 31;
  const int l16  = lane & 15;
  const int half = lane >> 4;             // 0: lanes 0-15, 1: lanes 16-31
  const int rowBase = blockIdx.x << 4;

  // stage 16x128 A tile into LDS as float4 (optionally through exact GELU)
  for (int i = tid; i < 16 * (HID / 4); i += blockDim.x) {
    int r = i >> 5, c4 = (i & 31) << 2;   // HID/4 == 32 chunks per row
    int gr = rowBase + r;
    float4 a = make_float4(0.f, 0.f, 0.f, 0.f);
    if (gr < M) a = *(const float4*)(A + (size_t)gr * HID + c4);
    if (geluIn) {
      a.x = gelu_exact(a.x); a.y = gelu_exact(a.y);
      a.z = gelu_exact(a.z); a.w = gelu_exact(a.w);
    }
    *(float4*)(&As[r * 132 + c4]) = a;    // (r*132 + c4) % 4 == 0 -> 16B aligned
  }
  __syncthreads();

  const int colBase = wave << 4;
  const int col = colBase + l16;
  // rows kk = k + 2*half and kk+1; single base, constant offsets k*NC / k*NC+NC
  const float* __restrict__ Bp = B + (size_t)(half << 1) * NC + col;
  const float* __restrict__ Ap = &As[l16 * 132 + (half << 1)];

  v8f acc = {};
#pragma unroll
  for (int k = 0; k < HID; k += 4) {
    // A 16x4 fragment: lanes 0-15 hold K=k,k+1; lanes 16-31 hold K=k+2,k+3
    v2f a, b;
    a.x = Ap[k];
    a.y = Ap[k + 1];
    // B 4x16 fragment: row K striped across lanes, same K-halving as A
    b.x = Bp[k * NC];
    b.y = Bp[k * NC + NC];
    acc = __builtin_amdgcn_wmma_f32_16x16x4_f32(false, a, false, b,
                                                (short)0, acc, false, false);
  }

  float beta = 1.0f, onemb = 0.0f;
  if (skipP) {
    float s = *skipP;
    beta = 1.0f / (1.0f + expf(-s));
    onemb = 1.0f - beta;
  }
  const float bcol = bias ? bias[col] : 0.0f;

  // C/D layout: VGPR j -> row = 8*half + j, col = colBase + (lane&15)
#pragma unroll
  for (int j = 0; j < 8; ++j) {
    int row = rowBase + (half << 3) + j;
    if (row < M) {
      float v = acc[j] + bcol;
      if (reluOut) v = fmaxf(v, 0.0f);
      if (resid) v = beta * v + onemb * resid[(size_t)row * HID + col];
      C[(size_t)row * NC + col] = v;
    }
  }
}

// ---------------------------------------------------------------------------
// Fold per-head relation matrix into projection weight/bias:
//   Weff[:, h*16:(h+1)*16] = W[:, h*16:(h+1)*16] @ rel[h];  beff likewise.
// One block per (l,t,h); rel laid out (L,R,H,16,16) with r == t here.
// ---------------------------------------------------------------------------
__global__ __launch_bounds__(256) void fold_rel_kernel(
    const float* __restrict__ W, const float* __restrict__ b,
    const float* __restrict__ rel, float* __restrict__ Weff,
    float* __restrict__ beff)
{
  __shared__ float Rs[HD * HD];
  const int blk = blockIdx.x;             // (l*T + t)*H + h
  const int h  = blk & 7;
  const int lt = blk >> 3;
  const float* Wp = W + (size_t)lt * HID * HID + h * HD;
  const float* bp = b + (size_t)lt * HID + h * HD;
  const float* rp = rel + (size_t)blk * HD * HD;
  float* Wo = Weff + (size_t)lt * HID * HID + h * HD;
  float* bo = beff + (size_t)lt * HID + h * HD;

  const int tid = threadIdx.x;
  if (tid < HD * HD) Rs[tid] = rp[tid];
  __syncthreads();

  for (int o = tid; o < HID * HD; o += blockDim.x) {
    int row = o >> 4, j = o & 15;
    const float* wr = Wp + (size_t)row * HID;
    float s = 0.0f;
#pragma unroll
    for (int d = 0; d < HD; ++d) s += wr[d] * Rs[d * HD + j];
    Wo[(size_t)row * HID + j] = s;
  }
  if (tid < HD) {
    float s = 0.0f;
#pragma unroll
    for (int d = 0; d < HD; ++d) s += bp[d] * Rs[d * HD + tid];
    bo[tid] = s;
  }
}

// ---------------------------------------------------------------------------
// Edge phase: one thread per (edge, head).
// ---------------------------------------------------------------------------
__global__ void edge_score_kernel(const float* __restrict__ q,
    const float* __restrict__ kt, const int* __restrict__ src,
    const int* __restrict__ dst, const float* __restrict__ prel,
    float* __restrict__ scores, float* __restrict__ mmax, int E_)
{
  int idx = blockIdx.x * blockDim.x + threadIdx.x;
  if (idx >= E_ * NH) return;
  int e = idx >> 3, h = idx & 7;
  int sN = src[e], dN = dst[e];
  const float4* qp = (const float4*)(q  + (size_t)dN * HID + h * HD);
  const float4* kp = (const float4*)(kt + (size_t)sN * HID + h * HD);
  float acc = 0.0f;
#pragma unroll
  for (int i = 0; i < 4; ++i) {
    float4 qv = qp[i], kv = kp[i];
    acc += qv.x * kv.x + qv.y * kv.y + qv.z * kv.z + qv.w * kv.w;
  }
  float sc = acc * prel[h] * 0.25f;       // 1/sqrt(D), D=16
  scores[idx] = sc;
  atomicMaxFloat(&mmax[dN * NH + h], sc);
}

__global__ void edge_exp_kernel(const int* __restrict__ dst,
    const float* __restrict__ mmax, float* __restrict__ scores,
    float* __restrict__ ssum, int E_)
{
  int idx = blockIdx.x * blockDim.x + threadIdx.x;
  if (idx >= E_ * NH) return;
  int e = idx >> 3, h = idx & 7;
  int dN = dst[e];
  float ev = expf(scores[idx] - mmax[dN * NH + h]);
  scores[idx] = ev;
  atomicAdd(&ssum[dN * NH + h], ev);
}

__global__ void edge_scatter_kernel(const float* __restrict__ vt,
    const int* __restrict__ src, const int* __restrict__ dst,
    const float* __restrict__ scores, const float* __restrict__ ssum,
    float* __restrict__ agg, int E_)
{
  int idx = blockIdx.x * blockDim.x + threadIdx.x;
  if (idx >= E_ * NH) return;
  int e = idx >> 3, h = idx & 7;
  int sN = src[e], dN = dst[e];
  float alpha = scores[idx] / (ssum[dN * NH + h] + 1e-16f);
  const float* vp = vt + (size_t)sN * HID + h * HD;
  float* ap = agg + (size_t)dN * HID + h * HD;
#pragma unroll
  for (int i = 0; i < HD; ++i) atomicAdd(ap + i, alpha * vp[i]);
}

// ---------------------------------------------------------------------------

extern "C" void kernel_launch(void* const* d_in, const int* in_sizes, int n_in,
                              void* d_out, int out_size, void* d_ws, size_t ws_size,
                              hipStream_t stream)
{
  const float* x_author = (const float*)d_in[0];
  const float* x_paper  = (const float*)d_in[1];
  const int*   e_wr     = (const int*)d_in[2];
  const int*   e_rv     = (const int*)d_in[3];
  const float* W_in     = (const float*)d_in[4];
  const float* b_in     = (const float*)d_in[5];
  const float* Wq       = (const float*)d_in[6];
  const float* bq       = (const float*)d_in[7];
  const float* Wk       = (const float*)d_in[8];
  const float* bk       = (const float*)d_in[9];
  const float* Wv       = (const float*)d_in[10];
  const float* bv       = (const float*)d_in[11];
  const float* Wa       = (const float*)d_in[12];
  const float* ba       = (const float*)d_in[13];
  const float* skip     = (const float*)d_in[14];
  const float* a_rel    = (const float*)d_in[15];
  const float* m_rel    = (const float*)d_in[16];
  const float* p_rel    = (const float*)d_in[17];
  const float* W_out    = (const float*)d_in[18];
  const float* b_out    = (const float*)d_in[19];

  const int Nn   = in_sizes[0] / HID;   // 25000
  const int Ee   = in_sizes[2] / 2;     // 250000
  const int OUTC = in_sizes[19];        // 64
  const int L = 2, T = 2;

  float* ws = (float*)d_ws;
  const size_t NF = (size_t)Nn * HID;
  float* qb[2]  = { ws + 0 * NF,  ws + 1 * NF };
  float* kb[2]  = { ws + 2 * NF,  ws + 3 * NF };
  float* vb[2]  = { ws + 4 * NF,  ws + 5 * NF };
  float* agg[2] = { ws + 6 * NF,  ws + 7 * NF };
  float* xs[2]  = { ws + 8 * NF,  ws + 9 * NF };
  float* nx[2]  = { ws + 10 * NF, ws + 11 * NF };
  float* scores = ws + 12 * NF;
  float* mmax   = scores + (size_t)Ee * NH;
  float* ssum   = mmax + (size_t)Nn * NH;
  float* Wkeff  = ssum + (size_t)Nn * NH;
  float* Wveff  = Wkeff + (size_t)L * T * HID * HID;
  float* bkeff  = Wveff + (size_t)L * T * HID * HID;
  float* bveff  = bkeff + (size_t)L * T * HID;

  auto gemm = [&](const float* A, const float* B, const float* bias,
                  const float* resid, const float* skipP, float* C,
                  int Ncols, int geluIn, int reluOut) {
    dim3 grid((unsigned)((Nn + 15) / 16));
    if (Ncols == 128)
      gemm_wmma_kernel<128><<<grid, 256, 0, stream>>>(A, B, bias, resid, skipP,
                                                      C, Nn, geluIn, reluOut);
    else
      gemm_wmma_kernel<64><<<grid, 128, 0, stream>>>(A, B, bias, resid, skipP,
                                                     C, Nn, geluIn, reluOut);
  };
  auto fill = [&](float* p, float v, size_t n) {
    fill_kernel<<<(unsigned)((n + 255) / 256), 256, 0, stream>>>(p, v, (int)n);
  };

  // Fold a_rel into Wk/bk and m_rel into Wv/bv (relation r == src type t).
  fold_rel_kernel<<<L * T * NH, 256, 0, stream>>>(Wk, bk, a_rel, Wkeff, bkeff);
  fold_rel_kernel<<<L * T * NH, 256, 0, stream>>>(Wv, bv, m_rel, Wveff, bveff);

  // Input projection + ReLU
  gemm(x_author, W_in,             b_in,       nullptr, nullptr, xs[0], HID, 0, 1);
  gemm(x_paper,  W_in + HID * HID, b_in + HID, nullptr, nullptr, xs[1], HID, 0, 1);

  const int EH = Ee * NH;
  const unsigned egrid = (unsigned)((EH + 255) / 256);

  for (int l = 0; l < L; ++l) {
    for (int t = 0; t < T; ++t) {
      const size_t wOff = (size_t)(l * T + t) * HID * HID;
      const size_t bOff = (size_t)(l * T + t) * HID;
      gemm(xs[t], Wq + wOff,    bq + bOff,    nullptr, nullptr, qb[t], HID, 0, 0);
      gemm(xs[t], Wkeff + wOff, bkeff + bOff, nullptr, nullptr, kb[t], HID, 0, 0);
      gemm(xs[t], Wveff + wOff, bveff + bOff, nullptr, nullptr, vb[t], HID, 0, 0);
      fill(agg[t], 0.0f, NF);
    }
    for (int r = 0; r < 2; ++r) {
      const int s = r, t = 1 - r;                // (0->1 writes), (1->0 rev)
      const int* ei  = (r == 0) ? e_wr : e_rv;
      const int* src = ei;
      const int* dst = ei + Ee;
      fill(mmax, -INFINITY, (size_t)Nn * NH);
      fill(ssum, 0.0f, (size_t)Nn * NH);
      const float* pr = p_rel + (size_t)(l * 2 + r) * NH;
      edge_score_kernel<<<egrid, 256, 0, stream>>>(qb[t], kb[s], src, dst, pr,
                                                   scores, mmax, Ee);
      edge_exp_kernel<<<egrid, 256, 0, stream>>>(dst, mmax, scores, ssum, Ee);
      edge_scatter_kernel<<<egrid, 256, 0, stream>>>(vb[s], src, dst, scores,
                                                     ssum, agg[t], Ee);
    }
    for (int t = 0; t < T; ++t) {
      const size_t wOff = (size_t)(l * T + t) * HID * HID;
      const size_t bOff = (size_t)(l * T + t) * HID;
      // gelu(agg) @ Wa + ba, then sigma(skip)-gated residual with xs[t]
      gemm(agg[t], Wa + wOff, ba + bOff, xs[t], skip + (l * T + t), nx[t],
           HID, 1, 0);
    }
    float* t0 = xs[0]; float* t1 = xs[1];
    xs[0] = nx[0]; xs[1] = nx[1];
    nx[0] = t0;    nx[1] = t1;
  }

  float* out = (float*)d_out;
  gemm(xs[0], W_out, b_out, nullptr, nullptr, out,                      OUTC, 0, 0);
  gemm(xs[1], W_out, b_out, nullptr, nullptr, out + (size_t)Nn * OUTC, OUTC, 0, 0);
}